// LevinsonDurbin_47502338294563
// MI455X (gfx1250) — compile-verified
//
#include <hip/hip_runtime.h>
#include <stdint.h>
#include <math.h>

// Levinson-Durbin for batched Toeplitz systems, MI455X (gfx1250).
// One thread = one batch element. Data staged through LDS with the CDNA5
// async global<->LDS DMA path (ASYNCcnt) for coalesced b128 traffic.
//
// Roofline: 52 MB total traffic @ 23.3 TB/s => ~2.3 us floor; compute is
// O(M^2) per element (~0.6 KFLOP) => ~0.16 GFLOP total. The reference's
// batched 24x24 inverse (what WMMA could have fed) is eliminated
// algorithmically; remaining work is per-lane sequential, so the gfx1250
// win is the async DMA staging path, not matrix cores.

#define LPC_M   24          // order
#define NR      25          // floats per element (r0..r24) == output floats per element
#define TPB     256         // 8 wave32s per block
#define FLOATS_PER_BLOCK (TPB * NR)        // 6400 floats = 25600 bytes
#define CHUNKS_B128      (FLOATS_PER_BLOCK / 4)  // 1600 16-byte chunks
#define CHUNK_ITERS      ((CHUNKS_B128 + TPB - 1) / TPB)  // 7

// Fast reciprocal: v_rcp_f32 (~1e-7 rel) + 1 Newton step -> full f32 accuracy,
// 3 VALU ops instead of the ~10-op IEEE division sequence. Removes ~150 ops
// from the serial recursion chain (23 divides).
__device__ __forceinline__ float fast_rcp(float x) {
    float r = __builtin_amdgcn_rcpf(x);
    r = r * __builtin_fmaf(-x, r, 2.0f);
    return r;
}

__global__ __launch_bounds__(TPB) void
LevinsonDurbin_47502338294563_kernel(const float* __restrict__ r,
                                     float* __restrict__ out)
{
    __shared__ float smem[FLOATS_PER_BLOCK];

    const int tid = threadIdx.x;
    // Low 32 bits of the flat LDS-aperture address == wave-relative LDS byte
    // offset (ISA 10.2); that is exactly what the
    // async DMA's VDST/VSRC operand wants.
    const unsigned lds_base = (unsigned)(uintptr_t)(void*)smem;

    // ---------------- stage in: global -> LDS via async DMA ----------------
    const float* gin = r + (size_t)blockIdx.x * FLOATS_PER_BLOCK;
    #pragma unroll
    for (int i = 0; i < CHUNK_ITERS; ++i) {
        int idx = i * TPB + tid;
        if (idx < CHUNKS_B128) {
            unsigned voff  = (unsigned)idx * 16u;      // global byte offset (GVS mode)
            unsigned laddr = lds_base + voff;          // LDS byte address
            asm volatile("global_load_async_to_lds_b128 %0, %1, %2 offset:0"
                         :: "v"(laddr), "v"(voff), "s"(gin)
                         : "memory");
        }
    }
    asm volatile("s_wait_asynccnt 0" ::: "memory");
    __syncthreads();

    // ------------- pull this element's 25 lags into registers -------------
    // stride-25 dwords across lanes: gcd(25,64)==1 -> bank-conflict free.
    float rr[NR];
    #pragma unroll
    for (int i = 0; i < NR; ++i) rr[i] = smem[tid * NR + i];

    // ---------------- Levinson-Durbin recursion, order 24 ------------------
    // Solves R a = -r1 for symmetric Toeplitz R; E_m = r0 + sum a_i r_i.
    float a[LPC_M];
    float E = rr[0];
    a[0] = -rr[1] * fast_rcp(E);
    E = E * (1.0f - a[0] * a[0]);

    #pragma unroll
    for (int m = 2; m <= LPC_M; ++m) {
        // acc = r[m] + sum_{j=0}^{m-2} a[j] * r[m-1-j]
        float acc = rr[m];
        #pragma unroll
        for (int j = 0; j <= m - 2; ++j)
            acc += a[j] * rr[m - 1 - j];

        float k = -acc * fast_rcp(E);

        // symmetric in-place update: a[j] += k * a[m-2-j]
        #pragma unroll
        for (int lo = 0; lo < (m - 1) / 2; ++lo) {
            int hi = m - 2 - lo;
            float t1 = a[lo], t2 = a[hi];
            a[lo] = t1 + k * t2;
            a[hi] = t2 + k * t1;
        }
        if (((m - 1) & 1) != 0) {
            int mid = (m - 2) / 2;
            a[mid] = a[mid] + k * a[mid];
        }
        a[m - 1] = k;
        E = E * (1.0f - k * k);
    }

    float K = sqrtf(E);   // == sqrt(r0 + r1·a) by the Levinson error invariant

    // -------- write [K, a0..a23] back to this thread's own LDS row ---------
    smem[tid * NR + 0] = K;
    #pragma unroll
    for (int i = 0; i < LPC_M; ++i) smem[tid * NR + 1 + i] = a[i];
    __syncthreads();

    // ---------------- stage out: LDS -> global via async DMA ---------------
    float* gout = out + (size_t)blockIdx.x * FLOATS_PER_BLOCK;
    #pragma unroll
    for (int i = 0; i < CHUNK_ITERS; ++i) {
        int idx = i * TPB + tid;
        if (idx < CHUNKS_B128) {
            unsigned voff  = (unsigned)idx * 16u;
            unsigned laddr = lds_base + voff;
            asm volatile("global_store_async_from_lds_b128 %0, %1, %2 offset:0"
                         :: "v"(voff), "v"(laddr), "s"(gout)
                         : "memory");
        }
    }
    asm volatile("s_wait_asynccnt 0" ::: "memory");
}

extern "C" void kernel_launch(void* const* d_in, const int* in_sizes, int n_in,
                              void* d_out, int out_size, void* d_ws, size_t ws_size,
                              hipStream_t stream)
{
    const float* r  = (const float*)d_in[0];
    float*      out = (float*)d_out;

    // in_sizes[0] = BATCH * 25; BATCH = 262144 -> exactly 1024 full blocks.
    int batch  = in_sizes[0] / NR;
    int blocks = batch / TPB;

    LevinsonDurbin_47502338294563_kernel<<<blocks, TPB, 0, stream>>>(r, out);
}